// DenseEquivariantIrrep_59115929862267
// MI455X (gfx1250) — compile-verified
//
#include <hip/hip_runtime.h>
#include <hip/hip_bf16.h>

// ---------------------------------------------------------------------------
// DenseEquivariantIrrep on gfx1250 (MI455X), f32 WMMA 16x16x4 path.
// B=32768, C=32, F=32, N_SYMM=48; irreps: 4x d=1, 2x d=2, 4x d=3.
// All WMMA "B" operands stored K-pair interleaved so a B-fragment is one
// aligned ds_load_b64: B[(k>>1)*pitch + 2n + (k&1)], pitch chosen so the two
// half-wave K-rows hit disjoint LDS bank halves (pitch % 64 == 32 floats).
// ---------------------------------------------------------------------------

typedef __attribute__((ext_vector_type(2))) float v2f;
typedef __attribute__((ext_vector_type(8))) float v8f;

#define WMMA_F32(a, b, c)                                                      \
  __builtin_amdgcn_wmma_f32_16x16x4_f32(false, (a), false, (b), (short)0, (c), \
                                        false, false)

// ---------------- workspace layout (floats) ----------------
// [0,2304)      fwd  interleaved: (s>>1)*96 + col*2 + (s&1)
// [2304,4608)   inv  interleaved: (s2>>1)*96 + s'*2 + (s2&1)
// [4608, +59392) Bk per-irrep B matrices [(c*d+r) x (f*d+q)], interleaved:
//   d1 x4: pitch 96,  16 pair-rows -> 1536/inst   @ 0,1536,3072,4608
//   d2 x2: pitch 160, 32 pair-rows -> 5120/inst   @ 6144,11264
//   d3 x4: pitch 224, 48 pair-rows -> 10752/inst  @ 16384,27136,37888,48640
#define WS_FWD 0
#define WS_INV 2304
#define WS_BK 4608

// ---------------- setup kernel 1: build fwd / inv (interleaved) ------------
__global__ void build_ft_kernel(const float* __restrict__ d1,
                                const float* __restrict__ d2,
                                const float* __restrict__ d3,
                                float* __restrict__ ws) {
  int e = blockIdx.x * blockDim.x + threadIdx.x;
  if (e >= 48 * 48) return;
  int s = e / 48, col = e % 48;
  float v, scale;
  if (col < 4) {  // d=1 irreps [4,48,1,1]
    v = d1[col * 48 + s];
    scale = 1.0f / 48.0f;
  } else if (col < 12) {  // d=2 irreps [2,48,2,2]
    int t = col - 4;
    int n = t >> 2, pr = t & 3;
    v = d2[(n * 48 + s) * 4 + pr];
    scale = 2.0f / 48.0f;
  } else {  // d=3 irreps [4,48,3,3]
    int t = col - 12;
    int n = t / 9, pr = t % 9;
    v = d3[(n * 48 + s) * 9 + pr];
    scale = 3.0f / 48.0f;
  }
  // fwd: K = s, N = col
  ws[WS_FWD + (s >> 1) * 96 + col * 2 + (s & 1)] = v;
  // inv: K = col (s2), N = s (s')
  ws[WS_INV + (col >> 1) * 96 + s * 2 + (col & 1)] = v * scale;
}

// ---------------- setup kernel 2: build Bk (FT'd kernel, interleaved) ------
__global__ void build_bk_kernel(const float* __restrict__ kern,
                                float* __restrict__ ws) {
  int e = blockIdx.x * blockDim.x + threadIdx.x;
  if (e >= 49152) return;
  int c, r, f, q, off, addr;
  if (e < 4096) {  // d=1: row=c (K), colB=f (N)
    int inst = e >> 10, w = e & 1023;
    int row = w >> 5, colB = w & 31;
    c = row; r = 0;
    f = colB; q = 0;
    off = inst;
    addr = inst * 1536 + (row >> 1) * 96 + colB * 2 + (row & 1);
  } else if (e < 12288) {  // d=2
    int e2 = e - 4096;
    int inst = e2 >> 12, w = e2 & 4095;
    int row = w >> 6, colB = w & 63;
    c = row >> 1; r = row & 1;
    f = colB >> 1; q = colB & 1;
    off = 4 + inst * 4 + r * 2 + q;
    addr = 6144 + inst * 5120 + (row >> 1) * 160 + colB * 2 + (row & 1);
  } else {  // d=3
    int e3 = e - 12288;
    int inst = e3 / 9216, w = e3 % 9216;
    int row = w / 96, colB = w % 96;
    c = row / 3; r = row % 3;
    f = colB / 3; q = colB % 3;
    off = 12 + inst * 9 + r * 3 + q;
    addr = 16384 + inst * 10752 + (row >> 1) * 224 + colB * 2 + (row & 1);
  }
  const float* kp = kern + (f * 32 + c) * 48;
  // fwd is interleaved: element (s, off) at (s>>1)*96 + off*2 + (s&1)
  float acc = 0.0f;
#pragma unroll 4
  for (int s = 0; s < 48; ++s)
    acc += kp[s] * ws[WS_FWD + (s >> 1) * 96 + off * 2 + (s & 1)];
  ws[WS_BK + addr] = acc;
}

// ---------------- LDS layout (floats) ----------------
// sXhP: per-irrep A matrices [(b*d+p) x (c*d+r)], lda = 32d+2:
//   d1 x4: 16x34  (544)  @ 0,544,1088,1632
//   d2 x2: 32x66  (2112) @ 2176,4288
//   d3 x4: 48x98  (4704) @ 6400,11104,15808,20512   total 25216
// sYh : assembled [(b*32+f) x 48], ld=50 -> 25600
// sBk : staged interleaved B slice, max 10752 (d3)
#define LDS_XHP 25216
#define LDS_YH 25600
#define LDS_BK 10752

// ------------- stage-2 tile compute, templated on irrep dim D --------------
template <int D>
__device__ __forceinline__ void stage2_tile(const float* __restrict__ sA,
                                            const float* __restrict__ sB,
                                            float* __restrict__ sYh, int mt,
                                            int nt, int colOff, int hl,
                                            int ll) {
  const int lda = 32 * D + 2;
  const int pb = (D == 1) ? 96 : (D == 2) ? 160 : 224;  // interleaved pitch
  v8f acc = {};
#pragma unroll
  for (int k = 0; k < 8 * D; ++k) {
    int kk = 4 * k + 2 * hl;
    v2f a = *(const v2f*)(sA + (mt * 16 + ll) * lda + kk);
    v2f b = *(const v2f*)(sB + (2 * k + hl) * pb + (nt * 16 + ll) * 2);
    acc = WMMA_F32(a, b, acc);
  }
#pragma unroll
  for (int v = 0; v < 8; ++v) {
    int M = mt * 16 + v + 8 * hl;  // = b_local*D + p
    int N = nt * 16 + ll;          // = f*D + q
    int bl = M / D, p = M - bl * D;
    int f = N / D, q = N - f * D;
    sYh[(bl * 32 + f) * 50 + colOff + p * D + q] = acc[v];
  }
}

// ---------------- fused main kernel: one WG per 16 rows of B ---------------
__global__ __launch_bounds__(256) void equi_dense_fused(
    const float* __restrict__ x, const float* __restrict__ bias,
    const float* __restrict__ ws, float* __restrict__ y) {
  __shared__ float sXhP[LDS_XHP];
  __shared__ float sYh[LDS_YH];
  __shared__ float sBk[LDS_BK];
  __shared__ float sFwd[2304];
  __shared__ float sInv[2304];
  __shared__ float sBias[32];

  const int tid = threadIdx.x;
  const int wave = tid >> 5;
  const int lane = tid & 31;
  const int hl = lane >> 4;  // half-wave select
  const int ll = lane & 15;
  const int b0 = blockIdx.x * 16;

  for (int i = tid; i < 2304; i += 256) {
    sFwd[i] = ws[WS_FWD + i];
    sInv[i] = ws[WS_INV + i];
  }
  if (tid < 32) sBias[tid] = bias[tid];
  __syncthreads();

  // ---------------- Stage 1: Xh = X @ fwd, scattered per-irrep -------------
  // rows (b_local,c): 512 -> 32 M-tiles; N = 48 -> 3 tiles; K = 48 -> 12 steps
  for (int mt = wave; mt < 32; mt += 8) {
    v2f a[12];
    const float* xp = x + (size_t)(b0 * 32 + mt * 16 + ll) * 48 + 2 * hl;
#pragma unroll
    for (int k = 0; k < 12; ++k) a[k] = *(const v2f*)(xp + 4 * k);
#pragma unroll
    for (int nt = 0; nt < 3; ++nt) {
      v8f acc = {};
#pragma unroll
      for (int k = 0; k < 12; ++k) {
        v2f b = *(const v2f*)(sFwd + (2 * k + hl) * 96 + (nt * 16 + ll) * 2);
        acc = WMMA_F32(a[k], b, acc);
      }
      // scatter: column s2 fixed per lane; rows vary with v
      int s2 = nt * 16 + ll;
      int dd, pr, abase;
      if (s2 < 4) {
        dd = 1; abase = s2 * 544; pr = 0;
      } else if (s2 < 12) {
        int t = s2 - 4;
        dd = 2; abase = 2176 + (t >> 2) * 2112; pr = t & 3;
      } else {
        int t = s2 - 12;
        dd = 3; abase = 6400 + (t / 9) * 4704; pr = t % 9;
      }
      int p = pr / dd, r = pr - p * dd;
      int lda = 32 * dd + 2;
#pragma unroll
      for (int v = 0; v < 8; ++v) {
        int row = mt * 16 + v + 8 * hl;  // (b_local, c)
        int bl = row >> 5, c = row & 31;
        sXhP[abase + (bl * dd + p) * lda + c * dd + r] = acc[v];
      }
    }
  }
  __syncthreads();

  // ---------------- Stage 2: per-irrep block GEMMs -------------------------
  // group d=1 (4 instances, Bk slice 6144 floats, 8 output tiles)
  for (int i = tid * 4; i < 6144; i += 1024)
    *(float4*)(sBk + i) = *(const float4*)(ws + WS_BK + i);
  __syncthreads();
  {
    int t = wave;  // 8 tiles, one per wave
    int inst = t >> 1, nt = t & 1;
    stage2_tile<1>(sXhP + inst * 544, sBk + inst * 1536, sYh, 0, nt, inst, hl,
                   ll);
  }
  __syncthreads();

  // group d=2 (2 instances, Bk slice 10240 floats, 16 output tiles)
  for (int i = tid * 4; i < 10240; i += 1024)
    *(float4*)(sBk + i) = *(const float4*)(ws + WS_BK + 6144 + i);
  __syncthreads();
  for (int t = wave; t < 16; t += 8) {
    int inst = t >> 3, sub = t & 7;
    int mt = sub >> 2, nt = sub & 3;
    stage2_tile<2>(sXhP + 2176 + inst * 2112, sBk + inst * 5120, sYh, mt, nt,
                   4 + inst * 4, hl, ll);
  }

  // group d=3 (4 instances, Bk slice 10752 floats each, 18 tiles each)
  for (int inst = 0; inst < 4; ++inst) {
    __syncthreads();
    const float* src = ws + WS_BK + 16384 + inst * 10752;
    for (int i = tid * 4; i < 10752; i += 1024)
      *(float4*)(sBk + i) = *(const float4*)(src + i);
    __syncthreads();
    for (int t = wave; t < 18; t += 8) {
      int mt = t / 6, nt = t % 6;
      stage2_tile<3>(sXhP + 6400 + inst * 4704, sBk, sYh, mt, nt,
                     12 + inst * 9, hl, ll);
    }
  }
  __syncthreads();

  // ---------------- Stage 3: Y = Yh @ inv + bias ---------------------------
  for (int mt = wave; mt < 32; mt += 8) {
    v2f a[12];
#pragma unroll
    for (int k = 0; k < 12; ++k) {
      int kk = 4 * k + 2 * hl;
      a[k] = *(const v2f*)(sYh + (mt * 16 + ll) * 50 + kk);
    }
#pragma unroll
    for (int nt = 0; nt < 3; ++nt) {
      v8f acc = {};
#pragma unroll
      for (int k = 0; k < 12; ++k) {
        v2f b = *(const v2f*)(sInv + (2 * k + hl) * 96 + (nt * 16 + ll) * 2);
        acc = WMMA_F32(a[k], b, acc);
      }
#pragma unroll
      for (int v = 0; v < 8; ++v) {
        int row = mt * 16 + v + 8 * hl;  // (b_local, f)
        int f = row & 31;
        y[(size_t)(b0 * 32 + row) * 48 + nt * 16 + ll] = acc[v] + sBias[f];
      }
    }
  }
}

// ---------------------------------------------------------------------------
extern "C" void kernel_launch(void* const* d_in, const int* in_sizes, int n_in,
                              void* d_out, int out_size, void* d_ws,
                              size_t ws_size, hipStream_t stream) {
  const float* x = (const float*)d_in[0];     // [B,C,48]
  const float* kern = (const float*)d_in[1];  // [F,C,48]
  const float* bias = (const float*)d_in[2];  // [F]
  const float* i1 = (const float*)d_in[3];    // [4,48,1,1]
  const float* i2 = (const float*)d_in[4];    // [2,48,2,2]
  const float* i3 = (const float*)d_in[5];    // [4,48,3,3]
  float* y = (float*)d_out;                   // [B,F,48]
  float* ws = (float*)d_ws;

  build_ft_kernel<<<9, 256, 0, stream>>>(i1, i2, i3, ws);
  build_bk_kernel<<<192, 256, 0, stream>>>(kern, ws);
  equi_dense_fused<<<32768 / 16, 256, 0, stream>>>(x, bias, ws, y);
}